// SimpleHGN_9689446220173
// MI455X (gfx1250) — compile-verified
//
#include <hip/hip_runtime.h>

typedef __attribute__((ext_vector_type(2))) float v2f;
typedef __attribute__((ext_vector_type(8))) float v8f;

#define HEADS 4
#define DHEAD 64
#define FDIM 256   // H*D = F_IN = 256
#define NEG_SLOPE 0.2f

// ---------- monotonic float<->uint encoding for atomic max ----------
__device__ __forceinline__ unsigned fkey(float f) {
    unsigned u = __float_as_uint(f);
    return (u & 0x80000000u) ? ~u : (u | 0x80000000u);
}
__device__ __forceinline__ float funkey(unsigned k) {
    unsigned u = (k & 0x80000000u) ? (k & 0x7FFFFFFFu) : ~k;
    return __uint_as_float(u);
}

// ---------------------------------------------------------------
// GEMM: Y[N,256] = X[N,256] @ W[256,256], fp32 via V_WMMA_F32_16X16X4_F32
// Block = 256 threads (8 waves). Each block: 16 rows x 128 cols.
// Wave w -> 16x16 tile at col (blockIdx.y*128 + w*16). A staged in LDS.
// ---------------------------------------------------------------
__global__ __launch_bounds__(256) void gemm_wmma_f32(
    const float* __restrict__ X, const float* __restrict__ W,
    float* __restrict__ Y, int N)
{
    __shared__ float As[16 * FDIM];              // 16 KB
    const int rowBase = blockIdx.x * 16;

    // cooperative load of 16x256 A strip (4096 floats), float4 coalesced
    const float4* src = (const float4*)(X + (size_t)rowBase * FDIM);
    float4* dsts = (float4*)As;
    for (int i = threadIdx.x; i < 16 * FDIM / 4; i += 256) dsts[i] = src[i];
    __syncthreads();

    const int wave  = threadIdx.x >> 5;
    const int lane  = threadIdx.x & 31;
    const int lhalf = lane >> 4;                 // 0: K0/K1 half, 1: K2/K3 half
    const int lidx  = lane & 15;
    const int colBase = blockIdx.y * 128 + wave * 16;
    const int col = colBase + lidx;

    v8f acc = {0.f, 0.f, 0.f, 0.f, 0.f, 0.f, 0.f, 0.f};
    #pragma unroll 4
    for (int k0 = 0; k0 < FDIM; k0 += 4) {
        const int ka = k0 + lhalf * 2;
        // A 16x4 fp32 layout: lane<16 -> row=lidx,{K0,K1}; lane>=16 -> {K2,K3}
        v2f a;
        a.x = As[lidx * FDIM + ka];
        a.y = As[lidx * FDIM + ka + 1];
        // B 4x16 fp32 layout (mirror): lane<16 -> col=lidx rows {K0,K1}; etc.
        v2f b;
        b.x = W[(size_t)ka * FDIM + col];
        b.y = W[(size_t)(ka + 1) * FDIM + col];
        acc = __builtin_amdgcn_wmma_f32_16x16x4_f32(
            false, a, false, b, (short)0, acc, false, false);
    }
    // D layout: VGPR r -> row (r + lhalf*8), N = lidx
    float* dst = Y + (size_t)(rowBase + lhalf * 8) * FDIM + col;
    #pragma unroll
    for (int r = 0; r < 8; ++r) dst[(size_t)r * FDIM] = acc[r];
}

// ---------------------------------------------------------------
// Per-(node, head) dot: out[n,h] = sum_d a[h,d] * Hmat[n, h*64+d]
// 8 waves/block -> 2 nodes/block, one wave per (node, head)
// ---------------------------------------------------------------
__global__ __launch_bounds__(256) void rowdot(
    const float* __restrict__ Hmat, const float* __restrict__ a,
    float* __restrict__ out, int N)
{
    const int wave = threadIdx.x >> 5;
    const int lane = threadIdx.x & 31;
    const int node = blockIdx.x * 2 + (wave >> 2);
    const int hd   = wave & 3;
    if (node >= N) return;
    const float* row = Hmat + (size_t)node * FDIM + hd * DHEAD;
    const float* av  = a + hd * DHEAD;
    float p = row[lane] * av[lane] + row[lane + 32] * av[lane + 32];
    #pragma unroll
    for (int off = 16; off; off >>= 1) p += __shfl_down(p, off, 32);
    if (lane == 0) out[node * HEADS + hd] = p;
}

// ---------------------------------------------------------------
// h_e[t,h] = sum_d a_e[h,d] * (edge_emb @ W_e)[t, h*64+d], 5 types
// ---------------------------------------------------------------
__global__ __launch_bounds__(256) void he_kernel(
    const float* __restrict__ emb, const float* __restrict__ We,
    const float* __restrict__ ae, float* __restrict__ he)
{
    __shared__ float Es[5 * FDIM];
    const int j = threadIdx.x;
    for (int t = 0; t < 5; ++t) {
        float s = 0.f;
        #pragma unroll 8
        for (int k = 0; k < 64; ++k) s += emb[t * 64 + k] * We[k * FDIM + j];
        Es[t * FDIM + j] = s;
    }
    __syncthreads();
    if (j < 20) {
        const int t = j >> 2, hd = j & 3;
        float s = 0.f;
        for (int d = 0; d < 64; ++d) s += ae[hd * 64 + d] * Es[t * FDIM + hd * 64 + d];
        he[j] = s;
    }
}

// ---------------------------------------------------------------
// Pass A: score = leaky_relu(hl[head] + hr[tail] + he[type]); seg-max
// ---------------------------------------------------------------
__global__ __launch_bounds__(256) void edge_score(
    const int* __restrict__ head, const int* __restrict__ tail,
    const int* __restrict__ etype, const float* __restrict__ hl,
    const float* __restrict__ hr, const float* __restrict__ he,
    float* __restrict__ score, unsigned* __restrict__ mkey, int E)
{
    const int e = blockIdx.x * 256 + threadIdx.x;
    if (e >= E) return;
    const int hI = head[e], tI = tail[e], ty = etype[e];
    const float4 l = *(const float4*)(hl + (size_t)hI * 4);
    const float4 r = *(const float4*)(hr + (size_t)tI * 4);
    const float4 x = *(const float4*)(he + (size_t)ty * 4);
    float4 s;
    s.x = l.x + r.x + x.x; s.y = l.y + r.y + x.y;
    s.z = l.z + r.z + x.z; s.w = l.w + r.w + x.w;
    s.x = s.x > 0.f ? s.x : NEG_SLOPE * s.x;
    s.y = s.y > 0.f ? s.y : NEG_SLOPE * s.y;
    s.z = s.z > 0.f ? s.z : NEG_SLOPE * s.z;
    s.w = s.w > 0.f ? s.w : NEG_SLOPE * s.w;
    *(float4*)(score + (size_t)e * 4) = s;
    unsigned* mb = mkey + (size_t)hI * 4;
    atomicMax(mb + 0, fkey(s.x));
    atomicMax(mb + 1, fkey(s.y));
    atomicMax(mb + 2, fkey(s.z));
    atomicMax(mb + 3, fkey(s.w));
}

// ---------------------------------------------------------------
// Pass B: ex = exp(score - m[head]); z[head] += ex (in-place over score)
// ---------------------------------------------------------------
__global__ __launch_bounds__(256) void edge_exp(
    const int* __restrict__ head, float* __restrict__ score,
    const unsigned* __restrict__ mkey, float* __restrict__ z, int E)
{
    const int e = blockIdx.x * 256 + threadIdx.x;
    if (e >= E) return;
    const int hI = head[e];
    float4 s = *(const float4*)(score + (size_t)e * 4);
    const unsigned* mb = mkey + (size_t)hI * 4;
    s.x = __expf(s.x - funkey(mb[0]));
    s.y = __expf(s.y - funkey(mb[1]));
    s.z = __expf(s.z - funkey(mb[2]));
    s.w = __expf(s.w - funkey(mb[3]));
    *(float4*)(score + (size_t)e * 4) = s;
    float* zb = z + (size_t)hI * 4;
    unsafeAtomicAdd(zb + 0, s.x);
    unsafeAtomicAdd(zb + 1, s.y);
    unsafeAtomicAdd(zb + 2, s.z);
    unsafeAtomicAdd(zb + 3, s.w);
}

// ---------------------------------------------------------------
// Pass C: acc[head,:] += alpha * h_tail[tail,:]  (one wave per edge,
// 8 coalesced channels per lane, hardware fp32 atomics)
// ---------------------------------------------------------------
__global__ __launch_bounds__(256) void edge_aggregate(
    const int* __restrict__ head, const int* __restrict__ tail,
    const float* __restrict__ ex, const float* __restrict__ z,
    const float* __restrict__ h_tail, float* __restrict__ acc, int E)
{
    const int wave = threadIdx.x >> 5;
    const int lane = threadIdx.x & 31;
    const int e = blockIdx.x * 8 + wave;
    if (e >= E) return;
    const int hI = head[e], tI = tail[e];
    const float4 exv = *(const float4*)(ex + (size_t)e * 4);
    const float4 zv  = *(const float4*)(z + (size_t)hI * 4);
    float alpha[4] = { exv.x / zv.x, exv.y / zv.y, exv.z / zv.z, exv.w / zv.w };
    const float* src = h_tail + (size_t)tI * FDIM;
    float* dst = acc + (size_t)hI * FDIM;
    #pragma unroll
    for (int i = 0; i < 8; ++i) {
        const int c = lane + i * 32;
        unsafeAtomicAdd(dst + c, alpha[i >> 1] * src[c]);
    }
}

// ---------------------------------------------------------------
// Final: out = elu(acc)
// ---------------------------------------------------------------
__global__ __launch_bounds__(256) void elu_kernel(
    const float* __restrict__ acc, float* __restrict__ out, int n)
{
    const int i = blockIdx.x * 256 + threadIdx.x;
    if (i < n) {
        const float x = acc[i];
        out[i] = x > 0.f ? x : (__expf(x) - 1.f);
    }
}

extern "C" void kernel_launch(void* const* d_in, const int* in_sizes, int n_in,
                              void* d_out, int out_size, void* d_ws, size_t ws_size,
                              hipStream_t stream) {
    (void)n_in; (void)out_size; (void)ws_size;
    const float* head_feat = (const float*)d_in[0];
    const float* tail_feat = (const float*)d_in[1];
    const int*   edge_list = (const int*)d_in[2];
    const int*   etype     = (const int*)d_in[3];
    const float* W         = (const float*)d_in[4];
    const float* W_e       = (const float*)d_in[5];
    const float* edge_emb  = (const float*)d_in[6];
    const float* a_l       = (const float*)d_in[7];
    const float* a_r       = (const float*)d_in[8];
    const float* a_e       = (const float*)d_in[9];

    const int N = in_sizes[0] / FDIM;     // 50000
    const int E = in_sizes[3];            // 800000
    const int* head_ind = edge_list;
    const int* tail_ind = edge_list + E;

    // workspace layout (256B aligned slabs)
    char* ws = (char*)d_ws;
    size_t off = 0;
    auto alloc = [&](size_t bytes) {
        void* p = ws + off;
        off += (bytes + 255) & ~(size_t)255;
        return p;
    };
    float*    hbuf   = (float*)alloc((size_t)N * FDIM * 4);  // h_head, later acc
    float*    h_tail = (float*)alloc((size_t)N * FDIM * 4);
    float*    score  = (float*)alloc((size_t)E * HEADS * 4); // then ex
    float*    hl     = (float*)alloc((size_t)N * HEADS * 4);
    float*    hr     = (float*)alloc((size_t)N * HEADS * 4);
    unsigned* mkey   = (unsigned*)alloc((size_t)N * HEADS * 4);
    float*    z      = (float*)alloc((size_t)N * HEADS * 4);
    float*    he     = (float*)alloc(32 * 4);

    // 1. fp32 WMMA GEMMs
    dim3 gGemm(N / 16, FDIM / 128);
    gemm_wmma_f32<<<gGemm, 256, 0, stream>>>(head_feat, W, hbuf, N);
    gemm_wmma_f32<<<gGemm, 256, 0, stream>>>(tail_feat, W, h_tail, N);

    // 2. attention logit reductions
    rowdot<<<(N + 1) / 2, 256, 0, stream>>>(hbuf,   a_l, hl, N);
    rowdot<<<(N + 1) / 2, 256, 0, stream>>>(h_tail, a_r, hr, N);
    he_kernel<<<1, 256, 0, stream>>>(edge_emb, W_e, a_e, he);

    // 3. segment softmax over edges
    hipMemsetAsync(mkey, 0, (size_t)N * HEADS * 4, stream);  // key 0 == -inf
    hipMemsetAsync(z,    0, (size_t)N * HEADS * 4, stream);
    edge_score<<<(E + 255) / 256, 256, 0, stream>>>(
        head_ind, tail_ind, etype, hl, hr, he, score, mkey, E);
    edge_exp<<<(E + 255) / 256, 256, 0, stream>>>(head_ind, score, mkey, z, E);

    // 4. weighted aggregation (hbuf reused as accumulator)
    hipMemsetAsync(hbuf, 0, (size_t)N * FDIM * 4, stream);
    edge_aggregate<<<(E + 7) / 8, 256, 0, stream>>>(
        head_ind, tail_ind, score, z, h_tail, hbuf, E);

    // 5. ELU -> output
    elu_kernel<<<((N * FDIM) + 255) / 256, 256, 0, stream>>>(
        hbuf, (float*)d_out, N * FDIM);
}